// MetaConv2d_update_54322746359868
// MI455X (gfx1250) — compile-verified
//
#include <hip/hip_runtime.h>

typedef __attribute__((ext_vector_type(2))) float v2f;
typedef __attribute__((ext_vector_type(8))) float v8f;

#define META 64
#define CIN  64
#define COUT 64
#define KW   3
#define SDIM 64
#define SOUT 62
#define NH   4096   /* CIN*META  */
#define NW2  192    /* COUT*KW   */
#define WBP  72     /* sWB row pitch: 2*72 % 64 == 16 -> half-waves hit disjoint banks */

__device__ __forceinline__ v8f wmma4(v2f a, v2f b, v8f c) {
  // V_WMMA_F32_16X16X4_F32 : D = A(16x4 f32) * B(4x16 f32) + C(16x16 f32)
  return __builtin_amdgcn_wmma_f32_16x16x4_f32(
      /*neg_a=*/false, a, /*neg_b=*/false, b,
      /*c_mod=*/(short)0, c, /*reuse_a=*/false, /*reuse_b=*/false);
}

// ---------------------------------------------------------------------------
// Kernel A: for a tile of 16 samples, H = mk @ w1^T + b1  ([16,64]x[64,4096])
//           and per-sample bias b = mk @ wb^T + bb.
// w1 (1 MB) is read once per workgroup -> stays L2 resident across the grid.
// H is stored block-tiled: H[blk][ntile(256)][gl(16)][m(16)] so the 8
// accumulator stores per round share one base address + immediate offsets
// and are fully coalesced.
// ---------------------------------------------------------------------------
__global__ __launch_bounds__(256) void hyper_stage1(
    const float* __restrict__ mk, const float* __restrict__ w1,
    const float* __restrict__ b1, const float* __restrict__ wb,
    const float* __restrict__ bb, float* __restrict__ Hout,
    float* __restrict__ Bias, int G)
{
  __shared__ float sMk[16 * 68];    // 16 samples x 64 meta (padded rows)
  __shared__ float sW1[128 * 68];   // 128 rows of w1 staged per round

  const int tid  = threadIdx.x;
  const int lane = tid & 31;
  const int wave = tid >> 5;
  const int gbase = blockIdx.x * 16;

  for (int i = tid; i < 16 * META; i += 256) {
    int gl = i >> 6, d = i & 63;
    int gg = gbase + gl; if (gg > G - 1) gg = G - 1;   // clamp loads only
    sMk[gl * 68 + d] = mk[(size_t)gg * META + d];
  }
  __syncthreads();

  // per-sample bias: 16*64 outputs, 4 per thread (plain VALU, tiny)
  for (int p = 0; p < 4; ++p) {
    int idx = tid + (p << 8);
    int gl = idx >> 6, o = idx & 63;
    float acc = bb[o];
    const float* wr = wb + o * META;
#pragma unroll 8
    for (int m2 = 0; m2 < META; ++m2) acc += sMk[gl * 68 + m2] * wr[m2];
    if (gbase + gl < G) Bias[(size_t)(gbase + gl) * COUT + o] = acc;
  }

  const int m  = lane & 15;
  const int kh = lane >> 4;

  // A fragments (mk tile) are reused across all 256 N-tiles: hoist all 16 k-steps
  v2f afrag[16];
#pragma unroll
  for (int ks = 0; ks < 16; ++ks) {
    afrag[ks][0] = sMk[m * 68 + ks * 4 + kh * 2 + 0];
    afrag[ks][1] = sMk[m * 68 + ks * 4 + kh * 2 + 1];
  }

  float* const hblk = Hout + (size_t)blockIdx.x * (256 * 256) + kh * 128 + m;

  // 4096 output columns = 256 tiles of 16; 8 tiles per round (one per wave)
  for (int round = 0; round < 32; ++round) {
    __syncthreads();
    const int rbase = round * 128;
    for (int i = tid; i < 128 * 16; i += 256) {     // stage 128 w1 rows (float4)
      int r = i >> 4, c4 = i & 15;
      float4 v = ((const float4*)(w1 + (size_t)(rbase + r) * META))[c4];
      float* dst = &sW1[r * 68 + c4 * 4];
      dst[0] = v.x; dst[1] = v.y; dst[2] = v.z; dst[3] = v.w;
    }
    // prefetch next round's w1 tile (32 KB = 256 cachelines, 1 per thread)
    if (round < 31)
      __builtin_prefetch(w1 + (size_t)(rbase + 128) * META + tid * 32, 0, 3);
    __syncthreads();

    const int ntile = round * 8 + wave;
    const int rl = wave * 16;
    v8f acc = {0.f, 0.f, 0.f, 0.f, 0.f, 0.f, 0.f, 0.f};
#pragma unroll
    for (int ks = 0; ks < 16; ++ks) {
      v2f bfrag;   // B[d][j] = w1[j][d]
      bfrag[0] = sW1[(rl + m) * 68 + ks * 4 + kh * 2 + 0];
      bfrag[1] = sW1[(rl + m) * 68 + ks * 4 + kh * 2 + 1];
      acc = wmma4(afrag[ks], bfrag, acc);
    }
    const float b1v = b1[ntile * 16 + m];
    float* hp = hblk + ntile * 256;                 // [ntile][gl=v+8kh][m]
#pragma unroll
    for (int v = 0; v < 8; ++v)
      hp[v * 16] = acc[v] + b1v;                    // imm-offset store clause
  }
}

// ---------------------------------------------------------------------------
// Kernel B: per sample g
//   1) W = H[g](64x64) @ w2^T + b2   (WMMA, result kept in LDS as sWB[r][o])
//   2) y[t,o] = sum_{r=(c,k)} x[g,t+k,c] * sWB[r][o] + bias     (WMMA, Kred=192)
// sWB is r-major so conv B-fragment ds loads are base + immediate offset.
// ---------------------------------------------------------------------------
__global__ __launch_bounds__(256) void hyper_conv(
    const float* __restrict__ x,  const float* __restrict__ w2,
    const float* __restrict__ b2, const float* __restrict__ H,
    const float* __restrict__ Bias, float* __restrict__ out)
{
  extern __shared__ float smem[];
  float* sH  = smem;                 // 64 x 66
  float* sW2 = sH  + 64 * 66;        // 192 x 66
  float* sWB = sW2 + 192 * 66;       // 192 x WBP : generated weights, r=(c*3+k) major
  float* sX  = sWB + 192 * WBP;      // 66 x 66   : x tile + 2 pad rows

  const int tid  = threadIdx.x;
  const int lane = tid & 31;
  const int wave = tid >> 5;
  const size_t g = blockIdx.x;

  // H is block-tiled: [g>>4][ntile=i>>4][gl=g&15][m=i&15]
  const size_t hbase = ((size_t)(g >> 4)) * (256 * 256) + (size_t)(g & 15) * 16;
  for (int i = tid; i < NH; i += 256) {
    sH[(i >> 6) * 66 + (i & 63)] = H[hbase + (size_t)(i >> 4) * 256 + (i & 15)];
    sX[(i >> 6) * 66 + (i & 63)] = x[g * (size_t)(SDIM * CIN) + i];
  }
  for (int i = tid; i < NW2 * META; i += 256)
    sW2[(i >> 6) * 66 + (i & 63)] = w2[i];
  if (tid < 128) sX[(64 + (tid >> 6)) * 66 + (tid & 63)] = 0.f;  // window overrun rows
  __syncthreads();

  const int m  = lane & 15;
  const int kh = lane >> 4;

  // ---- stage 1: W[c][n] = sum_d H[c][d]*w2[n][d] + b2[n]  (M=64,N=192,K=64)
  // wave -> fixed mt, 6 consecutive nt; A frags hoisted once per wave.
  {
    const int mt  = wave & 3;
    const int ntb = (wave >> 2) * 6;
    const int arow = (mt * 16 + m) * 66;
    v2f a2[16];
#pragma unroll
    for (int ks = 0; ks < 16; ++ks) {
      a2[ks][0] = sH[arow + ks * 4 + kh * 2 + 0];
      a2[ks][1] = sH[arow + ks * 4 + kh * 2 + 1];
    }
#pragma unroll
    for (int nti = 0; nti < 6; ++nti) {
      const int nt = ntb + nti;
      const int brow = (nt * 16 + m) * 66;
      v8f acc = {0.f, 0.f, 0.f, 0.f, 0.f, 0.f, 0.f, 0.f};
#pragma unroll
      for (int ks = 0; ks < 16; ++ks) {
        v2f b;
        b[0] = sW2[brow + ks * 4 + kh * 2 + 0];
        b[1] = sW2[brow + ks * 4 + kh * 2 + 1];
        acc = wmma4(a2[ks], b, acc);
      }
      // scatter into r-major sWB: element (c = mt*16+v+8kh, nn) -> r = c*3+nn%3, o = nn/3
      const int nn = nt * 16 + m;
      const int o  = nn / 3;
      const int kk = nn - 3 * o;
      const float b2v = b2[nn];
      float* wp = &sWB[((mt * 16 + 8 * kh) * 3 + kk) * WBP + o];
#pragma unroll
      for (int v = 0; v < 8; ++v)
        wp[v * 3 * WBP] = acc[v] + b2v;             // imm-offset ds stores
    }
  }
  __syncthreads();

  // ---- stage 2: conv as GEMM; A[t][r] = sX[t + r%3][r/3], B[r][o] = sWB[r][o]
  // wave -> fixed mt, two nt; each A frag feeds two WMMAs (independent acc chains).
  {
    const int mt  = wave & 3;
    const int ntb = (wave >> 2) * 2;
    const int o0  = (ntb + 0) * 16 + m;
    const int o1  = (ntb + 1) * 16 + m;
    const float* ap  = &sX[(mt * 16 + m) * 66];     // + (k*66 + c) selected by kh
    const float* bp0 = &sWB[2 * kh * WBP + o0];     // + (4ks+v)*WBP imm
    const float* bp1 = &sWB[2 * kh * WBP + o1];
    v8f acc0 = {0.f, 0.f, 0.f, 0.f, 0.f, 0.f, 0.f, 0.f};
    v8f acc1 = {0.f, 0.f, 0.f, 0.f, 0.f, 0.f, 0.f, 0.f};
#pragma unroll
    for (int ks = 0; ks < 48; ++ks) {
      v2f a, b0, b1;
#pragma unroll
      for (int v = 0; v < 2; ++v) {
        // A offset: two compile-time constants selected by per-lane kh (v_cndmask)
        const int r0 = 4 * ks + v;        // kh == 0
        const int r1 = 4 * ks + 2 + v;    // kh == 1
        const int c0 = r0 / 3, k0 = r0 - 3 * c0;
        const int c1 = r1 / 3, k1 = r1 - 3 * c1;
        a[v]  = ap[kh ? (k1 * 66 + c1) : (k0 * 66 + c0)];
        b0[v] = bp0[(4 * ks + v) * WBP];  // pure immediate offsets
        b1[v] = bp1[(4 * ks + v) * WBP];
      }
      acc0 = wmma4(a, b0, acc0);
      acc1 = wmma4(a, b1, acc1);
    }
    const float bo0 = Bias[g * COUT + o0];
    const float bo1 = Bias[g * COUT + o1];
    // stores: one base per output column, imm offsets v*COUT
    float* outp = out + ((size_t)g * SOUT + mt * 16 + 8 * kh) * COUT;
    float* op0 = outp + o0;
    float* op1 = outp + o1;
#pragma unroll
    for (int v = 0; v < 8; ++v) {
      const int tt = mt * 16 + v + 8 * kh;
      if (tt < SOUT) {
        op0[v * COUT] = acc0[v] + bo0;
        op1[v * COUT] = acc1[v] + bo1;
      }
    }
  }
}

extern "C" void kernel_launch(void* const* d_in, const int* in_sizes, int n_in,
                              void* d_out, int out_size, void* d_ws, size_t ws_size,
                              hipStream_t stream) {
  const float* mk = (const float*)d_in[0];
  const float* x  = (const float*)d_in[1];
  const float* w1 = (const float*)d_in[2];
  const float* b1 = (const float*)d_in[3];
  const float* w2 = (const float*)d_in[4];
  const float* b2 = (const float*)d_in[5];
  const float* wb = (const float*)d_in[6];
  const float* bb = (const float*)d_in[7];
  const int G = in_sizes[0] / META;              // 6624

  const int gridA = (G + 15) / 16;
  float* Hws  = (float*)d_ws;                    // gridA * 65536 floats (~108 MB)
  float* Bias = Hws + (size_t)gridA * (256 * 256);
  float* out  = (float*)d_out;

  hyper_stage1<<<gridA, 256, 0, stream>>>(mk, w1, b1, wb, bb, Hws, Bias, G);

  const size_t smemB =
      (size_t)(64 * 66 + 192 * 66 + 192 * WBP + 66 * 66) * sizeof(float);
  hyper_conv<<<G, 256, smemB, stream>>>(x, w2, b2, Hws, Bias, out);
}